// MOLAttn_12902081757715
// MI455X (gfx1250) — compile-verified
//
#include <hip/hip_runtime.h>

typedef __attribute__((ext_vector_type(2))) float v2f;
typedef __attribute__((ext_vector_type(8))) float v8f;

constexpr int Bn = 2048, Tn = 200, Dn = 512, An = 512;
constexpr int G3 = 3 * An;   // 1536
constexpr int H1 = 128;

static __device__ __forceinline__ float sigf(float x) {
  return 1.0f / (1.0f + __expf(-x));
}

// ---------------------------------------------------------------------------
// K1: fused gi = rnn_c @ W_ih^T + b_ih  (fp32 WMMA), then GRU activation -> h
// grid: B/16 workgroups x 256 threads (8 waves); dynamic LDS = 16*1536 floats
// ---------------------------------------------------------------------------
__global__ void k_gru(const float* __restrict__ rnn_c,
                      const float* __restrict__ W_ih,
                      const float* __restrict__ b_ih,
                      const float* __restrict__ b_hh,
                      float* __restrict__ h_out) {
  extern __shared__ float lds[];                 // [16][1536] = 96 KB
  const int tid  = threadIdx.x;
  const int wave = tid >> 5;
  const int lane = tid & 31;
  const int half = lane >> 4;                    // 0: K=0,1  1: K=2,3
  const int l16  = lane & 15;
  const int r0   = blockIdx.x * 16;              // batch row base

  for (int bt = wave; bt < G3 / 16; bt += 8) {   // 96 column tiles / 8 waves
    const int n0 = bt * 16;
    v8f c = {};
    // A tile: rnn_c rows r0..r0+15 ; lane l16 holds row M=l16, pair (2*half, 2*half+1)
    const float* arow = rnn_c + (size_t)(r0 + l16) * Dn + 2 * half;
    // B tile: W_ih^T => B[k][n] = W_ih[n0+n][k]; lane l16 holds col N=l16
    const float* brow = W_ih + (size_t)(n0 + l16) * Dn + 2 * half;
    for (int k0 = 0; k0 < Dn; k0 += 4) {
      v2f a = *(const v2f*)(arow + k0);
      v2f b = *(const v2f*)(brow + k0);
      c = __builtin_amdgcn_wmma_f32_16x16x4_f32(false, a, false, b,
                                                (short)0, c, false, false);
    }
    const float bias = b_ih[n0 + l16];
    #pragma unroll
    for (int v = 0; v < 8; ++v)                  // D elem (M=v+8*half, N=l16)
      lds[(v + 8 * half) * G3 + n0 + l16] = c[v] + bias;
  }
  __syncthreads();

  // GRU (h0 == 0 so gh == b_hh): r,z,n elementwise; h = (1-z)*n
  for (int idx = tid; idx < 16 * An; idx += 256) {
    const int row = idx >> 9;                    // /512
    const int a   = idx & (An - 1);
    const float r = sigf(lds[row * G3 + a] + b_hh[a]);
    const float z = sigf(lds[row * G3 + An + a] + b_hh[An + a]);
    const float n = tanhf(lds[row * G3 + 2 * An + a] + r * b_hh[2 * An + a]);
    h_out[(size_t)(r0 + row) * An + a] = (1.0f - z) * n;
  }
}

// ---------------------------------------------------------------------------
// K2: t1 = tanh(h @ W1^T + b1)   (B x 128, K = 512) via fp32 WMMA
// grid: 128 blocks x 256 threads; one wave per 16x16 tile (1024 tiles)
// ---------------------------------------------------------------------------
__global__ void k_mlp1(const float* __restrict__ h,
                       const float* __restrict__ W1,
                       const float* __restrict__ b1,
                       float* __restrict__ t1) {
  const int tid  = threadIdx.x;
  const int gid  = blockIdx.x * 8 + (tid >> 5);  // global wave id 0..1023
  const int lane = tid & 31;
  const int half = lane >> 4;
  const int l16  = lane & 15;
  const int m0 = (gid >> 3) * 16;
  const int n0 = (gid & 7) * 16;
  v8f c = {};
  const float* arow = h  + (size_t)(m0 + l16) * An + 2 * half;
  const float* brow = W1 + (size_t)(n0 + l16) * An + 2 * half;
  for (int k0 = 0; k0 < An; k0 += 4) {
    v2f a = *(const v2f*)(arow + k0);
    v2f b = *(const v2f*)(brow + k0);
    c = __builtin_amdgcn_wmma_f32_16x16x4_f32(false, a, false, b,
                                              (short)0, c, false, false);
  }
  const float bias = b1[n0 + l16];
  #pragma unroll
  for (int v = 0; v < 8; ++v)
    t1[(size_t)(m0 + v + 8 * half) * H1 + n0 + l16] = tanhf(c[v] + bias);
}

// ---------------------------------------------------------------------------
// K3: params = t1 @ W2^T + b2 (15 per row) -> MOL weights w[b,t] (masked)
// one wave per batch row; grid 256 x 256 (8 waves/block)
// ---------------------------------------------------------------------------
__global__ void k_mol(const float* __restrict__ t1,
                      const float* __restrict__ W2,
                      const float* __restrict__ b2,
                      const int* __restrict__ lens,
                      float* __restrict__ wout) {
  const int tid  = threadIdx.x;
  const int lane = tid & 31;
  const int row  = blockIdx.x * 8 + (tid >> 5);

  float p = 0.0f;
  if (lane < 15) {                               // 15 dot products of length 128
    p = b2[lane];
    const float* tr  = t1 + (size_t)row * H1;
    const float* w2r = W2 + lane * H1;
    for (int c = 0; c < H1; ++c) p = fmaf(tr[c], w2r[c], p);
  }
  float pj[15];
  #pragma unroll
  for (int j = 0; j < 15; ++j) pj[j] = __shfl(p, j, 32);  // wave32 broadcast

  float means[5], iscale[5], wk[5];
  float mx = -1e30f;
  #pragma unroll
  for (int k = 0; k < 5; ++k) {
    means[k]  = fminf(fmaxf(sigf(pj[k]) * 0.2f, 0.0f), (float)Tn);
    iscale[k] = __expf(-pj[5 + k]);              // 1/scale
    mx = fmaxf(mx, pj[10 + k]);
  }
  float se = 0.0f;
  #pragma unroll
  for (int k = 0; k < 5; ++k) { wk[k] = __expf(pj[10 + k] - mx); se += wk[k]; }
  const float ise = 1.0f / se;
  #pragma unroll
  for (int k = 0; k < 5; ++k) wk[k] *= ise;

  const int len = lens[row];
  for (int i = 0; i < 7; ++i) {
    const int t = i * 32 + lane;
    if (t < Tn) {
      float val = 1e-7f;
      #pragma unroll
      for (int k = 0; k < 5; ++k) {
        const float hi = sigf(((float)t + 0.5f - means[k]) * iscale[k]);
        const float lo = sigf(((float)t - 0.5f - means[k]) * iscale[k]);
        val = fmaf(wk[k], hi - lo, val);
      }
      if (t >= len) val = 0.0f;
      wout[(size_t)row * Tn + t] = val;
    }
  }
}

// ---------------------------------------------------------------------------
// K4: ctx[b,d] = sum_t w[b,t] * enc_z[b,t,d]  (bandwidth-bound, ~630 MB read)
// one 256-thread block per b; float2 coalesced loads; prefetch next rows
// ---------------------------------------------------------------------------
__global__ void k_ctx(const float* __restrict__ enc_z,
                      const int* __restrict__ lens,
                      const float* __restrict__ w_all,
                      float* __restrict__ ctx) {
  __shared__ float wl[Tn];
  const int b   = blockIdx.x;
  const int tid = threadIdx.x;
  if (tid < Tn) wl[tid] = w_all[(size_t)b * Tn + tid];
  __syncthreads();

  const int len = lens[b];
  const float2* ez = (const float2*)(enc_z + (size_t)b * Tn * Dn) + tid;
  float2 acc = {0.0f, 0.0f};
  for (int t = 0; t < len; ++t) {
    if (t + 8 < len) __builtin_prefetch(ez + (size_t)(t + 8) * (Dn / 2), 0, 0);
    const float  wt = wl[t];
    const float2 v  = ez[(size_t)t * (Dn / 2)];
    acc.x = fmaf(wt, v.x, acc.x);
    acc.y = fmaf(wt, v.y, acc.y);
  }
  float* o = ctx + (size_t)b * Dn + 2 * tid;
  o[0] = acc.x;
  o[1] = acc.y;
}

// ---------------------------------------------------------------------------
extern "C" void kernel_launch(void* const* d_in, const int* in_sizes, int n_in,
                              void* d_out, int out_size, void* d_ws, size_t ws_size,
                              hipStream_t stream) {
  const float* enc_z = (const float*)d_in[0];
  const int*   lens  = (const int*)  d_in[1];
  const float* rnn_c = (const float*)d_in[2];
  const float* W_ih  = (const float*)d_in[3];
  // d_in[4] = W_hh: mathematically unused since h0 == 0 (gh == b_hh)
  const float* b_ih  = (const float*)d_in[5];
  const float* b_hh  = (const float*)d_in[6];
  const float* W1    = (const float*)d_in[7];
  const float* b1    = (const float*)d_in[8];
  const float* W2    = (const float*)d_in[9];
  const float* b2    = (const float*)d_in[10];

  float* ctx  = (float*)d_out;                        // (B, 512)
  float* wout = (float*)d_out + (size_t)Bn * Dn;      // (B, 200)

  float* h  = (float*)d_ws;                           // B*512 floats (4 MB)
  float* t1 = h + (size_t)Bn * An;                    // B*128 floats (1 MB)

  k_gru <<<Bn / 16, 256, 16 * G3 * sizeof(float), stream>>>(rnn_c, W_ih, b_ih, b_hh, h);
  k_mlp1<<<(Bn / 16) * (H1 / 16) / 8, 256, 0, stream>>>(h, W1, b1, t1);
  k_mol <<<Bn / 8, 256, 0, stream>>>(t1, W2, b2, lens, wout);
  k_ctx <<<Bn, 256, 0, stream>>>(enc_z, lens, wout, ctx);
}